// NMSModule_36189394436620
// MI455X (gfx1250) — compile-verified
//
#include <hip/hip_runtime.h>
#include <hip/hip_bf16.h>
#include <stdint.h>

// ---------------------------------------------------------------------------
// NMS for gfx1250 (MI455X, wave32).
//  K1: single-block bitonic argsort (descending score, stable via index tiebreak)
//  K2: 16-row x N-col suppression bitmask; WMMA f32 16x16x4 computes the
//      separable (area_i + area_j) term; ballot_w32 packs the boolean tile.
//      Column tiles are software-pipelined + global_prefetch'd (WGP scope) ahead.
//  K3: single-wave sequential greedy scan; suppression word cached in a
//      register so dead rows cost only a bit test; barriers are single-wave
//      (S_NOP-level).
//  K4: scatter dets * keep back to original order.
// Workspace use (~8.6 MB): boxes(128K) area(32K) order(32K) sup(1K) mask(8M).
// ---------------------------------------------------------------------------

#define NBOX   8192
#define PITCH  (NBOX / 64)   // 128 u64 words per mask row

typedef __attribute__((ext_vector_type(2))) float v2f;
typedef __attribute__((ext_vector_type(8))) float v8f;

// ---------------- K1: bitonic sort ------------------------------------------
__global__ __launch_bounds__(1024) void nms_sort_kernel(const float* __restrict__ dets,
                                                        int* __restrict__ order,
                                                        float4* __restrict__ boxes,
                                                        float* __restrict__ area) {
    __shared__ float s_key[NBOX];
    __shared__ int   s_idx[NBOX];
    const int tid = threadIdx.x;

    for (int i = tid; i < NBOX; i += 1024) {
        s_key[i] = dets[i * 5 + 4];
        s_idx[i] = i;
    }
    __syncthreads();

    for (int k = 2; k <= NBOX; k <<= 1) {
        for (int j = k >> 1; j > 0; j >>= 1) {
            for (int i = tid; i < NBOX; i += 1024) {
                const int ixj = i ^ j;
                if (ixj > i) {
                    const bool desc = ((i & k) == 0);   // final pass => all descending
                    float a = s_key[i], b = s_key[ixj];
                    int   ia = s_idx[i], ib = s_idx[ixj];
                    // total order: higher score first; tie -> lower index first (stable)
                    const bool aFirst = (a > b) || ((a == b) && (ia < ib));
                    if (desc ? !aFirst : aFirst) {
                        s_key[i] = b;  s_key[ixj] = a;
                        s_idx[i] = ib; s_idx[ixj] = ia;
                    }
                }
            }
            __syncthreads();
        }
    }

    for (int i = tid; i < NBOX; i += 1024) {
        const int o = s_idx[i];
        order[i] = o;
        const float x1 = dets[o * 5 + 0];
        const float y1 = dets[o * 5 + 1];
        const float x2 = dets[o * 5 + 2];
        const float y2 = dets[o * 5 + 3];
        boxes[i] = make_float4(x1, y1, x2, y2);
        area[i]  = (x2 - x1 + 1.0f) * (y2 - y1 + 1.0f);
    }
}

// ---------------- K2: IoU bitmask, WMMA area-sum ----------------------------
// One wave per 16-row tile; sweeps all 512 column tiles, software-pipelined.
__global__ __launch_bounds__(32) void nms_mask_kernel(const float4* __restrict__ boxes,
                                                      const float* __restrict__ area,
                                                      const float* __restrict__ thr_p,
                                                      unsigned long long* __restrict__ mask) {
    const int lane = threadIdx.x;       // 0..31, wave32
    const int r0   = blockIdx.x * 16;   // first global (sorted) row of this tile
    const float t  = *thr_p;
    const float tp1 = 1.0f + t;

    __shared__ float4 rb[16];           // row boxes for this tile
    if (lane < 16) rb[lane] = boxes[r0 + lane];
    __syncthreads();

    // A fragment (16x4 f32): lanes 0-15 carry K0=area_row, K1=1; lanes 16-31 carry K2=K3=0.
    v2f afrag;
    afrag.x = (lane < 16) ? area[r0 + lane] : 0.0f;
    afrag.y = (lane < 16) ? 1.0f : 0.0f;

    const int nsub = lane & 15;         // column within tile this lane owns in C layout

    // software pipeline: preload tile 0
    float4 cb = boxes[nsub];
    float  ca = area[nsub];

    unsigned long long acc = 0ull;

    for (int ct = 0; ct < NBOX / 16; ++ct) {
        const int jcol = ct * 16 + nsub;

        // prefetch 8 tiles ahead into WGP-level cache (locality 3 -> near scope)
        int pf = jcol + 8 * 16;
        pf = (pf < NBOX) ? pf : jcol;
        __builtin_prefetch(&boxes[pf], 0, 3);

        // preload next tile into registers while computing this one
        int jn = jcol + 16;
        jn = (jn < NBOX) ? jn : nsub;
        const float4 cb_n = boxes[jn];
        const float  ca_n = area[jn];

        // B fragment (4x16 f32): lanes 0-15 carry K0=1 (VGPR0) and K1=area_col (VGPR1);
        // lanes 16-31 carry K2=K3=0.
        v2f bfrag;
        bfrag.x = (lane < 16) ? 1.0f : 0.0f;
        bfrag.y = (lane < 16) ? ca   : 0.0f;

        v8f czero = {};
        // S[k] = area_row(M) + area_col(N) for M = k + 8*(lane>=16), N = lane&15
        v8f S = __builtin_amdgcn_wmma_f32_16x16x4_f32(
            false, afrag, false, bfrag, (short)0, czero, false, false);

        unsigned int ball[8];
#pragma unroll
        for (int k = 0; k < 8; ++k) {
            const int M = k + ((lane >= 16) ? 8 : 0);
            const float4 r = rb[M];
            const float xx1 = fmaxf(r.x, cb.x);
            const float yy1 = fmaxf(r.y, cb.y);
            const float xx2 = fminf(r.z, cb.z);
            const float yy2 = fminf(r.w, cb.w);
            const float w = fmaxf(xx2 - xx1 + 1.0f, 0.0f);
            const float h = fmaxf(yy2 - yy1 + 1.0f, 0.0f);
            const float inter = w * h;
            // iou > t  <=>  inter*(1+t) > t*(area_i + area_j)
            const bool cond = (inter * tp1 > t * S[k]) && (jcol > (r0 + M));
            // ballot of C-VGPR k: bits[15:0] = row k, bits[31:16] = row k+8
            ball[k] = __builtin_amdgcn_ballot_w32(cond);
        }

        // lane r (r<16) gathers the 16-bit mask of row r
        unsigned int m16 = 0u;
#pragma unroll
        for (int k = 0; k < 8; ++k) {
            m16 |= (lane == k)     ? (ball[k] & 0xFFFFu) : 0u;
            m16 |= (lane == k + 8) ? (ball[k] >> 16)     : 0u;
        }

        acc |= (unsigned long long)m16 << (16 * (ct & 3));
        if ((ct & 3) == 3) {
            if (lane < 16)
                mask[(size_t)(r0 + lane) * PITCH + (ct >> 2)] = acc;
            acc = 0ull;
        }

        cb = cb_n;
        ca = ca_n;
    }
}

// ---------------- K3: sequential greedy scan (single wave) ------------------
__global__ __launch_bounds__(32) void nms_scan_kernel(const unsigned long long* __restrict__ mask,
                                                      unsigned long long* __restrict__ sup) {
    __shared__ unsigned long long rem[PITCH];
    const int tid = threadIdx.x;   // 32 lanes, 4 u64 words each
#pragma unroll
    for (int k = 0; k < 4; ++k) rem[tid * 4 + k] = 0ull;
    __syncthreads();               // single wave: counter waits + S_NOP

    for (int w = 0; w < PITCH; ++w) {
        unsigned long long cur = rem[w];           // uniform broadcast read
        for (int b = 0; b < 64; ++b) {
            const int i = (w << 6) + b;
            const int ipf = (i + 1 < NBOX) ? (i + 1) : i;
            __builtin_prefetch(&mask[(size_t)ipf * PITCH + tid * 4], 0, 3);
            if (((cur >> b) & 1ull) == 0ull) {     // row i alive -> apply its mask
                const unsigned long long* row = &mask[(size_t)i * PITCH];
#pragma unroll
                for (int k = 0; k < 4; ++k)
                    rem[tid * 4 + k] |= row[tid * 4 + k];
                __syncthreads();
                cur = rem[w];                      // refresh current word
            }
        }
    }
    __syncthreads();
#pragma unroll
    for (int k = 0; k < 4; ++k) sup[tid * 4 + k] = rem[tid * 4 + k];
}

// ---------------- K4: scatter output ----------------------------------------
__global__ __launch_bounds__(256) void nms_finalize_kernel(const float* __restrict__ dets,
                                                           const int* __restrict__ order,
                                                           const unsigned long long* __restrict__ sup,
                                                           float* __restrict__ out) {
    const int i = blockIdx.x * blockDim.x + threadIdx.x;  // sorted index
    if (i >= NBOX) return;
    const bool keep = ((sup[i >> 6] >> (i & 63)) & 1ull) == 0ull;
    const int o = order[i];
    const float f = keep ? 1.0f : 0.0f;
#pragma unroll
    for (int c = 0; c < 5; ++c)
        out[o * 5 + c] = dets[o * 5 + c] * f;
}

// ---------------- launcher ---------------------------------------------------
extern "C" void kernel_launch(void* const* d_in, const int* in_sizes, int n_in,
                              void* d_out, int out_size, void* d_ws, size_t ws_size,
                              hipStream_t stream) {
    const float* dets = (const float*)d_in[0];   // 8192 x 5
    const float* thr  = (const float*)d_in[1];   // scalar
    float* out = (float*)d_out;

    uintptr_t base = (uintptr_t)d_ws;
    float4* boxes = (float4*)(base);                              // 128 KB, 16B aligned
    float*  area  = (float*)(base + 131072);                      // 32 KB
    int*    order = (int*)(base + 131072 + 32768);                // 32 KB
    unsigned long long* sup  = (unsigned long long*)(base + 196608);   // 1 KB
    unsigned long long* mask = (unsigned long long*)(base + 197632);   // 8 MB

    nms_sort_kernel<<<1, 1024, 0, stream>>>(dets, order, boxes, area);
    nms_mask_kernel<<<NBOX / 16, 32, 0, stream>>>(boxes, area, thr, mask);
    nms_scan_kernel<<<1, 32, 0, stream>>>(mask, sup);
    nms_finalize_kernel<<<(NBOX + 255) / 256, 256, 0, stream>>>(dets, order, sup, out);
}